// ANN_16432544875395
// MI455X (gfx1250) — compile-verified
//
#include <hip/hip_runtime.h>
#include <hip/hip_bf16.h>

typedef __attribute__((ext_vector_type(2))) float v2f;
typedef __attribute__((ext_vector_type(8))) float v8f;

#define TPB 128
#define NKNOT 66
#define HF (1.0f / 65.0f)

// ---- compile-time Thomas factors for A = tridiag(H, 4H, H), size 64 ----
struct Tri { float cp[64]; float invd[64]; };
__host__ __device__ constexpr Tri make_tri() {
  Tri t{};
  float prev = 0.0f;
  for (int i = 0; i < 64; ++i) {
    float den = 4.0f * HF - HF * prev;
    t.invd[i] = 1.0f / den;
    t.cp[i]   = HF / den;
    prev = t.cp[i];
  }
  return t;
}
__constant__ Tri g_tri = make_tri();

// ---- kernel 0: query-point prep + range penalty ----
__global__ __launch_bounds__(128) void ann_prep(
    const float* __restrict__ pbi, const float* __restrict__ pdi,
    const float* __restrict__ pndi, const float* __restrict__ pmndi,
    int* __restrict__ wsIdx, float* __restrict__ wsF, float* __restrict__ wsPen) {
  __shared__ float red[128];
  const int t = threadIdx.x;
  float p = 0.0f;
  const bool valid = (t < 80);
  if (valid) {
    p = (t < 10) ? pbi[t] : (t < 30) ? pdi[t - 10] : (t < 50) ? pndi[t - 30] : pmndi[t - 50];
    int idx = (int)floorf(p * 65.0f);
    idx = idx < 0 ? 0 : (idx > 64 ? 64 : idx);
    wsIdx[t] = idx;
    wsF[t]   = p - (float)idx * HF;
  }
  red[t] = valid ? (fmaxf(-p, 0.0f) + fmaxf(p - 1.0f, 0.0f)) : 0.0f;
  __syncthreads();
  for (int s = 64; s > 0; s >>= 1) { if (t < s) red[t] += red[t + s]; __syncthreads(); }
  if (t == 0) *wsPen = red[0];  // alpha = 1
}

// ---- kernel 1: spline + WMMA MLP per 128-sample tile ----
__global__ __launch_bounds__(TPB) void ann_main(
    const float* __restrict__ x, const float* __restrict__ soc,
    const float* __restrict__ W1, const float* __restrict__ b1,
    const float* __restrict__ W2, const float* __restrict__ b2,
    const int* __restrict__ wsIdx, const float* __restrict__ wsF,
    float* __restrict__ out, int B, int ncols, float* __restrict__ partials) {
  // yM (TPB*133, spline phase) and outT (TPB*68, MLP phase) share one pool:
  // outputs are staged in registers across the phase boundary.
  __shared__ float pool[TPB * 133];
  __shared__ int   sIdx[80];
  __shared__ float sF[80];
  __shared__ float w1t[32 * 68];     // W1 transposed [n][k], padded, stride 68
  __shared__ float b1s[32];
  __shared__ float w2s[20];
  __shared__ float sb2;
  __shared__ float hid[TPB * 21];
  __shared__ float lossAcc;

  float* const yM   = pool;
  float* const outT = pool;

  const int t = threadIdx.x;
  const long long gbase = (long long)blockIdx.x * TPB;

  // phase 0: cooperative init (single pass per array, no intra-phase races)
  for (int i = t; i < 80; i += TPB) { sIdx[i] = wsIdx[i]; sF[i] = wsF[i]; }
  for (int i = t; i < 32 * 68; i += TPB) {
    int n = i / 68, k = i - n * 68;
    w1t[i] = (n < 20 && k < 50) ? W1[k * 20 + n] : 0.0f;
  }
  if (t < 32) b1s[t] = (t < 20) ? b1[t] : 0.0f;
  if (t < 20) w2s[t] = W2[t];
  if (t == 0) { sb2 = b2[0]; lossAcc = 0.0f; }
  // coalesced x tile load -> y region of yM
  for (int i = t; i < TPB * 66; i += TPB) {
    int r = i / 66, c = i - r * 66;
    long long s = gbase + r;
    yM[r * 133 + c] = (s < B) ? x[s * 66 + c] : 0.0f;
  }
  __syncthreads();

  // phase 1: per-thread Thomas solve + spline eval (results into registers)
  float ov[40];
  {
    float* Y = &yM[t * 133];
    float* M = Y + 66;
    float y0 = Y[0], y1 = Y[1];
    float dprev = 0.0f;
    #pragma unroll 1
    for (int i = 0; i < 64; ++i) {
      float y2 = Y[i + 2];
      float rhs = 6.0f * 65.0f * (y2 - 2.0f * y1 + y0);
      float dp = (rhs - HF * dprev) * g_tri.invd[i];
      M[i + 1] = dp; dprev = dp;
      y0 = y1; y1 = y2;
    }
    float mnext = M[64];
    #pragma unroll 1
    for (int i = 62; i >= 0; --i) {
      float mi = M[i + 1] - g_tri.cp[i] * mnext;
      M[i + 1] = mi; mnext = mi;
    }
    M[0] = 0.0f; M[65] = 0.0f;

    auto EV = [&](int j) -> float {
      int i = sIdx[j]; float f = sF[j];
      float a0 = Y[i], a1 = Y[i + 1], m0 = M[i], m1 = M[i + 1];
      float bc = (a1 - a0) * 65.0f - HF * (2.0f * m0 + m1) * (1.0f / 6.0f);
      float cc = 0.5f * m0;
      float dc = (m1 - m0) * (65.0f / 6.0f);
      return a0 + f * (bc + f * (cc + f * dc));
    };
    #pragma unroll
    for (int j = 0; j < 10; ++j) ov[j] = EV(j);
    #pragma unroll
    for (int k = 0; k < 10; ++k) ov[10 + k] = EV(10 + 2 * k) - EV(11 + 2 * k);
    #pragma unroll
    for (int k = 0; k < 10; ++k) { float a = EV(30 + 2 * k), bq = EV(31 + 2 * k); ov[20 + k] = (a - bq) / (a + bq); }
    #pragma unroll
    for (int k = 0; k < 10; ++k) {
      float a = EV(50 + 3 * k), bq = EV(51 + 3 * k), cq = EV(52 + 3 * k);
      ov[30 + k] = (a - bq) / (a + bq - 2.0f * cq);
    }
  }
  __syncthreads();   // all yM reads complete before pool is reused as outT

  // phase 1.5: publish outputs row into reused pool (K padded to 64, stride 68)
  {
    float* orow = &outT[t * 68];
    #pragma unroll
    for (int c = 0; c < 40; ++c) orow[c] = ov[c];
    #pragma unroll
    for (int c = 40; c < 64; ++c) orow[c] = 0.0f;
  }
  __syncthreads();

  // phase 2: WMMA f32 16x16x4: hidden = leaky_relu(outputs @ W1 + b1)
  {
    const int w = t >> 5, l = t & 31;
    const int lr = l & 15, lhi = l >> 4;   // lanes 16-31 carry K+2 / M+8 halves
    #pragma unroll
    for (int mt = 0; mt < 2; ++mt) {
      const int arow = w * 32 + mt * 16 + lr;
      #pragma unroll
      for (int nt = 0; nt < 2; ++nt) {
        const int n = nt * 16 + lr;
        v8f acc = {};
        #pragma unroll
        for (int ks = 0; ks < 16; ++ks) {
          const int k0 = ks * 4 + lhi * 2;
          v2f a;  a.x  = outT[arow * 68 + k0]; a.y  = outT[arow * 68 + k0 + 1];
          v2f bm; bm.x = w1t[n * 68 + k0];     bm.y = w1t[n * 68 + k0 + 1];
          acc = __builtin_amdgcn_wmma_f32_16x16x4_f32(
              false, a, false, bm, (short)0, acc, false, false);
        }
        const float bias = b1s[n];
        #pragma unroll
        for (int r = 0; r < 8; ++r) {
          float hv = acc[r] + bias;
          hv = (hv >= 0.0f) ? hv : 0.01f * hv;
          int row = w * 32 + mt * 16 + (lhi ? (r + 8) : r);
          if (n < 20) hid[row * 21 + n] = hv;
        }
      }
    }
  }
  __syncthreads();

  // phase 3: soc_hat, loss partial, coalesced outputs writeout
  {
    float acc2 = sb2;
    #pragma unroll
    for (int k = 0; k < 20; ++k) acc2 += hid[t * 21 + k] * w2s[k];
    long long s = gbase + t;
    float e2 = 0.0f;
    if (s < B) {
      out[s] = acc2;
      float e = acc2 - soc[s];
      e2 = e * e;
    }
    atomicAdd(&lossAcc, e2);
    float* gout = out + B + gbase * ncols;
    const int tot = TPB * ncols;
    for (int i = t; i < tot; i += TPB) {
      int r = i / ncols, c = i - r * ncols;   // ncols <= 64 (host-clamped)
      if (gbase + r < B) gout[i] = outT[r * 68 + c];
    }
  }
  __syncthreads();
  if (t == 0) partials[blockIdx.x] = lossAcc;
}

// ---- kernel 2: deterministic final loss reduction ----
__global__ __launch_bounds__(256) void ann_finish(
    const float* __restrict__ partials, int n,
    const float* __restrict__ pen, float* __restrict__ lossOut) {
  __shared__ float red[256];
  const int t = threadIdx.x;
  float a = 0.0f;
  for (int i = t; i < n; i += 256) a += partials[i];
  red[t] = a; __syncthreads();
  for (int s = 128; s > 0; s >>= 1) { if (t < s) red[t] += red[t + s]; __syncthreads(); }
  if (t == 0) *lossOut = red[0] + *pen;
}

extern "C" void kernel_launch(void* const* d_in, const int* in_sizes, int n_in,
                              void* d_out, int out_size, void* d_ws, size_t ws_size,
                              hipStream_t stream) {
  const float* x     = (const float*)d_in[0];
  const float* soc   = (const float*)d_in[1];
  const float* pbi   = (const float*)d_in[2];
  const float* pdi   = (const float*)d_in[3];
  const float* pndi  = (const float*)d_in[4];
  const float* pmndi = (const float*)d_in[5];
  const float* W1    = (const float*)d_in[6];
  const float* b1    = (const float*)d_in[7];
  const float* W2    = (const float*)d_in[8];
  const float* b2    = (const float*)d_in[9];

  const int B = in_sizes[1];
  long long nc = ((long long)out_size - B - 1) / B;
  int ncols = (int)nc;
  if (ncols < 40) ncols = 40;
  if (ncols > 64) ncols = 64;

  int*   wsIdx  = (int*)d_ws;
  float* wsF    = (float*)((char*)d_ws + 320);
  float* wsPen  = (float*)((char*)d_ws + 640);
  float* wsPart = (float*)((char*)d_ws + 1024);

  const int nblocks = (B + TPB - 1) / TPB;

  ann_prep<<<1, 128, 0, stream>>>(pbi, pdi, pndi, pmndi, wsIdx, wsF, wsPen);
  ann_main<<<nblocks, TPB, 0, stream>>>(x, soc, W1, b1, W2, b2, wsIdx, wsF,
                                        (float*)d_out, B, ncols, wsPart);
  ann_finish<<<1, 256, 0, stream>>>(wsPart, nblocks, wsPen,
                                    (float*)d_out + (size_t)B + (size_t)B * ncols);
}